// FpsPool_15891378995593
// MI455X (gfx1250) — compile-verified
//
#include <hip/hip_runtime.h>
#include <hip/hip_bf16.h>

typedef float v2f __attribute__((ext_vector_type(2)));
typedef float v8f __attribute__((ext_vector_type(8)));

#define KNBR 32
#define RADIUS2 0.0064f   // R=0.08
#define FPS_THREADS 1024

// ---------------------------------------------------------------------------
// Kernel 0: src coords -> SoA (x,y,z,|p|^2) so ball-query loads are coalesced
// ---------------------------------------------------------------------------
__global__ void prep_src_soa(const float* __restrict__ coord,
                             float* __restrict__ sx, float* __restrict__ sy,
                             float* __restrict__ sz, float* __restrict__ ssq, int N) {
    int i = blockIdx.x * blockDim.x + threadIdx.x;
    if (i < N) {
        float x = coord[3 * i + 0], y = coord[3 * i + 1], z = coord[3 * i + 2];
        sx[i] = x; sy[i] = y; sz[i] = z; ssq[i] = x * x + y * y + z * z;
    }
}

// ---------------------------------------------------------------------------
// Kernel 1: edge lists default to -1 (reference pads with -1)
// ---------------------------------------------------------------------------
__global__ void init_edges(float* __restrict__ es, float* __restrict__ ed, int n) {
    int i = blockIdx.x * blockDim.x + threadIdx.x;
    if (i < n) { es[i] = -1.0f; ed[i] = -1.0f; }
}

// ---------------------------------------------------------------------------
// Kernel 2: FPS, single workgroup, whole problem resident in 320KB CDNA5 LDS.
//   coords (240,000B) + min_d (80,000B) + reduction scratch  ~= 320,264B
// Argmax tracking is fused into the min-update pass; each of the 5000 serial
// steps costs one strided 20-elem register pass + two barriers.
// ---------------------------------------------------------------------------
__global__ void __launch_bounds__(FPS_THREADS) fps_kernel(
        const float* __restrict__ coord, int* __restrict__ idx, int N, int n_dst) {
    extern __shared__ float lds[];
    float* sx = lds;            // N
    float* sy = sx + N;         // N
    float* sz = sy + N;         // N
    float* md = sz + N;         // N  (min squared distance)
    float* rv = md + N;         // 32 (per-wave reduction values)
    int*   ri = (int*)(rv + 32);// 32 (per-wave reduction indices)
    int*   bc = ri + 32;        // 1  (broadcast slot)

    const int tid = threadIdx.x;

    for (int e = tid; e < N; e += FPS_THREADS) {
        sx[e] = coord[3 * e + 0];
        sy[e] = coord[3 * e + 1];
        sz[e] = coord[3 * e + 2];
    }
    __syncthreads();

    // d0 = distance to point 0; track local argmax as we go
    float x0 = sx[0], y0 = sy[0], z0 = sz[0];
    float lv = -1.0f; int li = 0x7fffffff;
    for (int e = tid; e < N; e += FPS_THREADS) {
        float dx = sx[e] - x0, dy = sy[e] - y0, dz = sz[e] - z0;
        float d = dx * dx + dy * dy + dz * dz;
        md[e] = d;
        if (d > lv) { lv = d; li = e; }   // strictly > keeps first index
    }
    if (tid == 0) idx[0] = 0;
    __syncthreads();

    for (int s = 1; s < n_dst; ++s) {
        // wave32 argmax reduction (first-index tie break), then cross-wave
        float v = lv; int i = li;
        #pragma unroll
        for (int off = 16; off; off >>= 1) {
            float ov = __shfl_xor(v, off, 32);
            int   oi = __shfl_xor(i, off, 32);
            if (ov > v || (ov == v && oi < i)) { v = ov; i = oi; }
        }
        if ((tid & 31) == 0) { rv[tid >> 5] = v; ri[tid >> 5] = i; }
        __syncthreads();
        if (tid < 32) {
            float v2 = rv[tid]; int i2 = ri[tid];
            #pragma unroll
            for (int off = 16; off; off >>= 1) {
                float ov = __shfl_xor(v2, off, 32);
                int   oi = __shfl_xor(i2, off, 32);
                if (ov > v2 || (ov == v2 && oi < i2)) { v2 = ov; i2 = oi; }
            }
            if (tid == 0) bc[0] = i2;
        }
        __syncthreads();
        int nxt = bc[0];
        if (tid == 0) idx[s] = nxt;

        float nx = sx[nxt], ny = sy[nxt], nz = sz[nxt];
        lv = -1.0f; li = 0x7fffffff;
        for (int e = tid; e < N; e += FPS_THREADS) {
            float dx = sx[e] - nx, dy = sy[e] - ny, dz = sz[e] - nz;
            float d  = dx * dx + dy * dy + dz * dz;
            float m  = md[e];
            float nm = d < m ? d : m;
            md[e] = nm;
            if (nm > lv) { lv = nm; li = e; }
        }
    }
}

// ---------------------------------------------------------------------------
// Kernel 3: gather dst coords -> SoA (padded to 128-row tiles), coord + batch out
// ---------------------------------------------------------------------------
__global__ void gather_dst(const float* __restrict__ coord, const int* __restrict__ batch,
                           const int* __restrict__ idx,
                           float* __restrict__ ax, float* __restrict__ ay,
                           float* __restrict__ az, float* __restrict__ asq,
                           float* __restrict__ o_coord, float* __restrict__ o_batch,
                           int n_dst, int n_pad) {
    int i = blockIdx.x * blockDim.x + threadIdx.x;
    if (i >= n_pad) return;
    if (i < n_dst) {
        int s = idx[i];
        float x = coord[3 * s + 0], y = coord[3 * s + 1], z = coord[3 * s + 2];
        ax[i] = x; ay[i] = y; az[i] = z; asq[i] = x * x + y * y + z * z;
        o_coord[3 * i + 0] = x; o_coord[3 * i + 1] = y; o_coord[3 * i + 2] = z;
        o_batch[i] = (float)batch[s];
    } else {
        // padding rows: force d2 huge so no neighbor is ever selected
        ax[i] = 0.0f; ay[i] = 0.0f; az[i] = 0.0f; asq[i] = 4e30f;
    }
}

// ---------------------------------------------------------------------------
// Kernel 4: ball query via V_WMMA_F32_16X16X4_F32.
//   D = A x B + C with A row = (ax,ay,az,1), B col = (-2bx,-2by,-2bz,|b|^2),
//   C = |a|^2  =>  D = full 16x16 tile of squared distances in one WMMA.
// B loads are branchless: per-lane base pointers + scale chosen once (lane
// half selects which SoA array the lane streams), and software-pipelined one
// chunk ahead so the L2 load latency retires under WMMA + selection work.
// Ordered first-K selection: per D-register ballot -> two 16-bit row masks in
// src-index order; per-lane popcount gives the write slot. Counts live in
// registers (uniform per half-wave). Early exit when all 16 rows hit K.
// One wave handles 16 dst rows; 256-thread block = 128 rows.
// ---------------------------------------------------------------------------
__global__ void __launch_bounds__(256) ball_query_wmma(
        const float* __restrict__ sx, const float* __restrict__ sy,
        const float* __restrict__ sz, const float* __restrict__ ssq,
        const float* __restrict__ ax, const float* __restrict__ ay,
        const float* __restrict__ az, const float* __restrict__ asq,
        float* __restrict__ o_esrc, float* __restrict__ o_edst,
        float* __restrict__ o_deg, int* __restrict__ nbr, int* __restrict__ deg,
        int N, int n_dst) {
    const int lane = threadIdx.x & 31;
    const int wave = threadIdx.x >> 5;
    const int half = lane >> 4;          // 0: rows r, K=0/1 side; 1: rows r+8, K=2/3 side
    const int li   = lane & 15;
    const int rowBase = blockIdx.x * 128 + wave * 16;

    // A-matrix (16x4 f32): lanes 0-15 hold (K0,K1)=(ax,ay) of row li,
    //                      lanes 16-31 hold (K2,K3)=(az,1) of row li.
    const int arow = rowBase + li;
    v2f A;
    A.x = half ? az[arow] : ax[arow];
    A.y = half ? 1.0f     : ay[arow];

    // Branchless per-lane B sources:
    //   lanes 0-15 : (K0,K2) = (-2*sx[col], -2*sz[col])
    //   lanes 16-31: (K1,K3) = (-2*sy[col],  1 * ssq[col])
    const float* __restrict__ b0base = half ? sy  : sx;
    const float* __restrict__ b1base = half ? ssq : sz;
    const float  b1scale = half ? 1.0f : -2.0f;

    // C init: VGPR r holds row rowBase + r (+8 for high lanes)
    v8f C;
    #pragma unroll
    for (int r = 0; r < 8; ++r) C[r] = asq[rowBase + (half << 3) + r];

    int cnt[8];
    #pragma unroll
    for (int r = 0; r < 8; ++r) cnt[r] = 0;

    // software pipeline: preload chunk 0
    v2f Bn;
    Bn.x = -2.0f    * b0base[li];
    Bn.y = b1scale  * b1base[li];

    for (int c = 0; c < N; c += 16) {
        const int col  = c + li;
        const bool colv = col < N;
        v2f B = Bn;

        // prefetch next chunk's B operands (uniform branch)
        int cn = c + 16;
        if (cn < N) {
            int ncol = cn + li;
            Bn.x = -2.0f   * b0base[ncol];
            Bn.y = b1scale * b1base[ncol];
        }

        v8f D = __builtin_amdgcn_wmma_f32_16x16x4_f32(
            false, A, false, B, (short)0, C, false, false);

        #pragma unroll
        for (int r = 0; r < 8; ++r) {
            bool pred = colv && (D[r] <= RADIUS2);
            unsigned mask  = __builtin_amdgcn_ballot_w32(pred);
            unsigned rmask = half ? (mask >> 16) : (mask & 0xFFFFu);
            int before = cnt[r];
            if (pred) {
                int pos  = before + __builtin_popcount(rmask & ((1u << li) - 1u));
                int grow = rowBase + (half << 3) + r;
                if (pos < KNBR && grow < n_dst) {
                    int e = grow * KNBR + pos;
                    o_esrc[e] = (float)col;
                    o_edst[e] = (float)grow;
                    nbr[e]    = col;
                }
            }
            int nc = before + __builtin_popcount(rmask);
            cnt[r] = nc > KNBR ? KNBR : nc;
        }
        bool done = true;
        #pragma unroll
        for (int r = 0; r < 8; ++r) done = done && (cnt[r] >= KNBR);
        if (__builtin_amdgcn_ballot_w32(done) == 0xFFFFFFFFu) break;
    }

    if (li == 0) {
        #pragma unroll
        for (int r = 0; r < 8; ++r) {
            int grow = rowBase + (half << 3) + r;
            if (grow < n_dst) { deg[grow] = cnt[r]; o_deg[grow] = (float)cnt[r]; }
        }
    }
}

// ---------------------------------------------------------------------------
// Kernel 5: scatter-mean of gathered src features (L2-resident gather),
// 2-way unrolled to keep two row-gathers in flight.
// ---------------------------------------------------------------------------
__global__ void feat_mean(const float* __restrict__ feat, const int* __restrict__ nbr,
                          const int* __restrict__ deg, float* __restrict__ out,
                          int NF) {
    int row = blockIdx.x;
    int f   = threadIdx.x;
    if (f >= NF) return;
    int d = deg[row];
    float acc0 = 0.0f, acc1 = 0.0f;
    int j = 0;
    for (; j + 1 < d; j += 2) {
        int n0 = nbr[row * KNBR + j];
        int n1 = nbr[row * KNBR + j + 1];
        acc0 += feat[(size_t)n0 * NF + f];
        acc1 += feat[(size_t)n1 * NF + f];
    }
    if (j < d) acc0 += feat[(size_t)nbr[row * KNBR + j] * NF + f];
    out[(size_t)row * NF + f] = (acc0 + acc1) / (float)(d > 0 ? d : 1);
}

// ---------------------------------------------------------------------------
extern "C" void kernel_launch(void* const* d_in, const int* in_sizes, int n_in,
                              void* d_out, int out_size, void* d_ws, size_t ws_size,
                              hipStream_t stream) {
    const float* coord = (const float*)d_in[0];   // [N,3]
    const float* feat  = (const float*)d_in[1];   // [N,128]
    const int*   batch = (const int*)d_in[2];     // [N]

    const int N     = in_sizes[0] / 3;
    const int NF    = in_sizes[1] / N;            // 128
    const int n_dst = N / 4;                      // RATIO = 0.25
    const int n_pad = ((n_dst + 127) / 128) * 128;

    // workspace layout
    float* sx  = (float*)d_ws;
    float* sy  = sx + N;
    float* sz  = sy + N;
    float* ssq = sz + N;
    float* ax  = ssq + N;
    float* ay  = ax + n_pad;
    float* az  = ay + n_pad;
    float* asq = az + n_pad;
    int* idx = (int*)(asq + n_pad);
    int* nbr = idx + n_pad;
    int* deg = nbr + (size_t)n_dst * KNBR;

    // output layout (concatenated in reference return order, all as f32)
    float* out     = (float*)d_out;
    float* o_coord = out;                                 // n_dst*3
    float* o_feat  = o_coord + (size_t)n_dst * 3;         // n_dst*NF
    float* o_esrc  = o_feat  + (size_t)n_dst * NF;        // n_dst*K
    float* o_edst  = o_esrc  + (size_t)n_dst * KNBR;      // n_dst*K
    float* o_deg   = o_edst  + (size_t)n_dst * KNBR;      // n_dst
    float* o_batch = o_deg   + n_dst;                     // n_dst

    prep_src_soa<<<(N + 255) / 256, 256, 0, stream>>>(coord, sx, sy, sz, ssq, N);

    init_edges<<<((n_dst * KNBR) + 255) / 256, 256, 0, stream>>>(o_esrc, o_edst, n_dst * KNBR);

    // 4*N floats of LDS + 64 reduction slots + 1 broadcast slot
    size_t fps_lds = (size_t)(4 * N) * sizeof(float) + 64 * sizeof(float) + 16;
    fps_kernel<<<1, FPS_THREADS, fps_lds, stream>>>(coord, idx, N, n_dst);

    gather_dst<<<(n_pad + 255) / 256, 256, 0, stream>>>(
        coord, batch, idx, ax, ay, az, asq, o_coord, o_batch, n_dst, n_pad);

    ball_query_wmma<<<n_pad / 128, 256, 0, stream>>>(
        sx, sy, sz, ssq, ax, ay, az, asq,
        o_esrc, o_edst, o_deg, nbr, deg, N, n_dst);

    feat_mean<<<n_dst, NF, 0, stream>>>(feat, nbr, deg, o_feat, NF);
}